// SSMLayer_22497038696759
// MI455X (gfx1250) — compile-verified
//
#include <hip/hip_runtime.h>
#include <hip/hip_bf16.h>
#include <math.h>

typedef float v2f __attribute__((ext_vector_type(2)));
typedef float v8f __attribute__((ext_vector_type(8)));
typedef int   v4i_vs __attribute__((vector_size(16)));   // matches builtin param

// Problem shape (fixed by the reference)
constexpr int BATCH = 4;
constexpr int SEQ   = 4096;
constexpr int HID   = 1024;
constexpr int SDIM  = 16;
constexpr int ROWS  = BATCH * SEQ;     // 16384  (row = b*SEQ + t)
constexpr int LCH   = 32;              // scan chunk length
constexpr int NCHB  = SEQ / LCH;       // 128 chunks per batch
constexpr int NCH   = BATCH * NCHB;    // 512 chunks total

// Workspace layout (float offsets); total ~3.3 MB of f32
constexpr int WS_BD    = 0;                           // B_d [16][1024]
constexpr int WS_P     = WS_BD + SDIM * HID;          // P_j = A_d^j, j=0..32
constexpr int WS_U     = WS_P + (LCH + 1) * 256;      // u [ROWS][16]
constexpr int WS_SL    = WS_U + ROWS * SDIM;          // local (zero-carry) states
constexpr int WS_CARRY = WS_SL + ROWS * SDIM;         // carry-in per chunk [NCH][16]
constexpr int WS_S     = WS_CARRY + NCH * SDIM;       // corrected states [ROWS][16]

// CDNA5 async global->LDS path (ASYNCcnt-tracked, no VGPR round trip)
#if defined(__gfx1250__) && __has_builtin(__builtin_amdgcn_global_load_async_to_lds_b128)
#define HAVE_ASYNC_LDS 1
#else
#define HAVE_ASYNC_LDS 0
#endif

#if __has_builtin(__builtin_amdgcn_s_wait_asynccnt)
#define WAIT_ASYNCCNT(n) __builtin_amdgcn_s_wait_asynccnt(n)
#else
#define WAIT_ASYNCCNT(n) asm volatile("s_wait_asynccnt " #n ::: "memory")
#endif

// ---------------------------------------------------------------------------
// f32 WMMA: D = A(16x4) * B(4x16) + C(16x16), full f32 precision.
// A layout (ISA 7.12.2): lane l -> M = l%16, VGPR v -> K = v + 2*(l/16)
// B layout (row-striped): lane l -> N = l%16, VGPR v -> K = v + 2*(l/16)
// C/D layout: lane l -> N = l%16, VGPR v -> M = v + 8*(l/16)
// ---------------------------------------------------------------------------
__device__ __forceinline__ v8f wmma_f32_16x16x4(v2f a, v2f b, v8f c) {
  return __builtin_amdgcn_wmma_f32_16x16x4_f32(false, a, false, b, (short)0, c,
                                               false, false);
}

// ===========================================================================
// Kernel 1: setup. dt, A_d = expm(A*dt) (Taylor 12 + 6 squarings),
// Ainv via Gauss-Jordan w/ partial pivoting, M = Ainv*(A_d - I),
// B_d = M*B, powers P_0..P_32.  One block, 256 threads (thread = elem (i,j)).
// ===========================================================================
__global__ __launch_bounds__(256) void k_setup(const float* __restrict__ A,
                                               const float* __restrict__ Bmat,
                                               const float* __restrict__ log_dt,
                                               float* __restrict__ ws) {
  __shared__ float red[256];
  __shared__ float Ms[256], term[256], acc[256], Ad[256];
  __shared__ float aug[16 * 33];
  __shared__ float fcol[16];
  __shared__ float Ainv[256], Mm[256], Pc[256];
  __shared__ float s_dt, s_inv;
  __shared__ int s_piv;

  const int tid = threadIdx.x;
  const int i = tid >> 4, j = tid & 15;

  // ---- dt = mean(exp(log_dt)) ----
  float s = 0.f;
  for (int q = tid; q < HID; q += 256) s += expf(log_dt[q]);
  red[tid] = s;
  __syncthreads();
  for (int o = 128; o > 0; o >>= 1) {
    if (tid < o) red[tid] += red[tid + o];
    __syncthreads();
  }
  if (tid == 0) s_dt = red[0] * (1.0f / (float)HID);
  __syncthreads();

  // ---- expm(A*dt): scale by 2^-6, Taylor 12, square 6x ----
  Ms[tid]   = A[tid] * (s_dt * (1.0f / 64.0f));
  term[tid] = (i == j) ? 1.f : 0.f;
  acc[tid]  = (i == j) ? 1.f : 0.f;
  __syncthreads();
  for (int it = 1; it <= 12; ++it) {
    float t2 = 0.f;
    for (int k = 0; k < 16; ++k) t2 += term[i * 16 + k] * Ms[k * 16 + j];
    __syncthreads();
    term[tid] = t2 / (float)it;
    acc[tid] += term[tid];
    __syncthreads();
  }
  for (int sq = 0; sq < 6; ++sq) {
    float t2 = 0.f;
    for (int k = 0; k < 16; ++k) t2 += acc[i * 16 + k] * acc[k * 16 + j];
    __syncthreads();
    acc[tid] = t2;
    __syncthreads();
  }
  Ad[tid] = acc[tid];
  __syncthreads();

  // ---- Gauss-Jordan inverse of A (augmented [A | I], partial pivoting) ----
  aug[i * 33 + j]      = A[tid];
  aug[i * 33 + 16 + j] = (i == j) ? 1.f : 0.f;
  __syncthreads();
  for (int k = 0; k < 16; ++k) {
    if (tid == 0) {
      int p = k;
      float best = fabsf(aug[k * 33 + k]);
      for (int r = k + 1; r < 16; ++r) {
        float v = fabsf(aug[r * 33 + k]);
        if (v > best) { best = v; p = r; }
      }
      s_piv = p;
    }
    __syncthreads();
    const int p = s_piv;
    if (p != k && tid < 32) {
      float t2 = aug[k * 33 + tid];
      aug[k * 33 + tid] = aug[p * 33 + tid];
      aug[p * 33 + tid] = t2;
    }
    __syncthreads();
    if (tid == 0) s_inv = 1.0f / aug[k * 33 + k];
    __syncthreads();
    if (tid < 32) aug[k * 33 + tid] *= s_inv;
    __syncthreads();
    if (tid < 16) fcol[tid] = aug[tid * 33 + k];
    __syncthreads();
    if (i != k) {
      aug[i * 33 + j]      -= fcol[i] * aug[k * 33 + j];
      aug[i * 33 + 16 + j] -= fcol[i] * aug[k * 33 + 16 + j];
    }
    __syncthreads();
  }
  Ainv[tid] = aug[i * 33 + 16 + j];
  __syncthreads();

  // ---- M = Ainv * (A_d - I) ----
  {
    float t2 = 0.f;
    for (int k = 0; k < 16; ++k) {
      float ad = Ad[k * 16 + j] - ((k == j) ? 1.f : 0.f);
      t2 += Ainv[i * 16 + k] * ad;
    }
    Mm[tid] = t2;
  }
  __syncthreads();

  // ---- B_d = M * B  ([16][1024]) ----
  for (int q = 0; q < (SDIM * HID) / 256; ++q) {
    int idx = q * 256 + tid;
    int n = idx >> 10, h = idx & (HID - 1);
    float t2 = 0.f;
    for (int k = 0; k < 16; ++k) t2 += Mm[n * 16 + k] * Bmat[k * HID + h];
    ws[WS_BD + idx] = t2;
  }

  // ---- powers P_0..P_LCH of A_d ----
  ws[WS_P + tid]       = (i == j) ? 1.f : 0.f;
  ws[WS_P + 256 + tid] = Ad[tid];
  Pc[tid] = Ad[tid];
  __syncthreads();
  for (int jp = 2; jp <= LCH; ++jp) {
    float t2 = 0.f;
    for (int k = 0; k < 16; ++k) t2 += Pc[i * 16 + k] * Ad[k * 16 + j];
    __syncthreads();
    Pc[tid] = t2;
    ws[WS_P + jp * 256 + tid] = t2;
    __syncthreads();
  }
}

// ===========================================================================
// Kernel 2: u = x @ B_d^T  (M=16384, K=1024, N=16) with f32 WMMA.
// Block = 256 threads = 8 waves; wave owns a 16-row tile (private LDS stage).
// Async path: GLOBAL_LOAD_ASYNC_TO_LDS_B128 double-buffered, ASYNCcnt-paced,
// zero barriers (DS ops are in-order in HW; explicit s_wait_dscnt guards the
// async-write-after-ds-read hazard since ASYNCcnt/DScnt are unordered).
// ===========================================================================
constexpr int KSW  = 32;   // K columns staged per step
constexpr int XPAD = 36;   // row pad: stride 36 mod 64 -> conflict-free banks

__global__ __launch_bounds__(256) void k_u_gemm(const float* __restrict__ x,
                                                float* __restrict__ ws) {
#if HAVE_ASYNC_LDS
  __shared__ float Xs[8][2][16 * XPAD];   // double buffer, 36.9 KB
#else
  __shared__ float Xs[8][1][16 * XPAD];
#endif
  const int tid = threadIdx.x;
  const int w = tid >> 5, lane = tid & 31;
  const int M = lane & 15, half = lane >> 4;
  const int rowbase = blockIdx.x * 128 + w * 16;
  const float* __restrict__ Bd = ws + WS_BD;

  v8f acc = {0.f, 0.f, 0.f, 0.f, 0.f, 0.f, 0.f, 0.f};

  auto compute = [&](int stq, int buf) {
    const int k0 = stq * KSW;
#pragma unroll
    for (int kk = 0; kk < KSW; kk += 4) {
      v2f a = *(const v2f*)&Xs[w][buf][M * XPAD + kk + 2 * half];
      v2f b = *(const v2f*)&Bd[M * HID + k0 + kk + 2 * half];
      acc = wmma_f32_16x16x4(a, b, acc);
    }
  };

#if HAVE_ASYNC_LDS
  auto stage_async = [&](int stq, int buf) {
    // previous ds reads of this buffer must have executed before async writes
    asm volatile("s_wait_dscnt 0x0" ::: "memory");
#pragma unroll
    for (int it = 0; it < 4; ++it) {   // 4 x B128 per lane = 16 rows x 32 cols
      const int r = it * 4 + (lane >> 3), c4 = lane & 7;
      __builtin_amdgcn_global_load_async_to_lds_b128(
          (__attribute__((address_space(1))) v4i_vs*)
              &x[(rowbase + r) * HID + stq * KSW + c4 * 4],
          (__attribute__((address_space(3))) v4i_vs*)
              &Xs[w][buf][r * XPAD + c4 * 4],
          0, 0);
    }
  };

  stage_async(0, 0);
  for (int st = 0; st < HID / KSW; ++st) {
    const int buf = st & 1;
    if (st + 1 < HID / KSW) {
      stage_async(st + 1, buf ^ 1);
      __builtin_prefetch(&x[(rowbase + M) * HID + (st + 1) * KSW], 0, 1);
      WAIT_ASYNCCNT(4);   // dones are in-order: only newest stage outstanding
    } else {
      WAIT_ASYNCCNT(0);
    }
    compute(st, buf);
  }
#else
  for (int st = 0; st < HID / KSW; ++st) {
    const int k0 = st * KSW;
#pragma unroll
    for (int it = 0; it < 4; ++it) {
      const int r = it * 4 + (lane >> 3), c4 = lane & 7;
      const float4 v = *(const float4*)&x[(rowbase + r) * HID + k0 + c4 * 4];
      *(float4*)&Xs[w][0][r * XPAD + c4 * 4] = v;
    }
    if (st + 1 < HID / KSW)
      __builtin_prefetch(&x[(rowbase + M) * HID + k0 + KSW], 0, 1);
    __syncthreads();
    compute(st, 0);
    __syncthreads();
  }
#endif

#pragma unroll
  for (int v = 0; v < 8; ++v)
    ws[WS_U + (rowbase + v + 8 * half) * SDIM + M] = acc[v];
}

// ===========================================================================
// Kernel 3: per-chunk local scan (zero carry-in): s_j = s_{j-1} A_d + u_j.
// 16 lanes per chunk, cross-lane via shfl(width 16); A_d column in registers.
// ===========================================================================
__global__ __launch_bounds__(256) void k_scan_local(float* __restrict__ ws) {
  const int tid = threadIdx.x;
  const int n = tid & 15;
  const int chunk = blockIdx.x * 16 + (tid >> 4);
  float Acol[16];
#pragma unroll
  for (int m = 0; m < 16; ++m) Acol[m] = ws[WS_P + 256 + m * 16 + n];
  float s = 0.f;
  const int base = chunk * LCH;
  for (int j2 = 0; j2 < LCH; ++j2) {
    float sn = ws[WS_U + (base + j2) * SDIM + n];
#pragma unroll
    for (int m = 0; m < 16; ++m) sn += __shfl(s, m, 16) * Acol[m];
    s = sn;
    ws[WS_SL + (base + j2) * SDIM + n] = s;
  }
}

// ===========================================================================
// Kernel 4: chunk carries: C_{c+1} = C_c * A_d^L + v_c. 128 small steps per
// batch (vs 4096 serial steps naively). Also writes final_state to out tail.
// ===========================================================================
__global__ __launch_bounds__(64) void k_carry(const float* __restrict__ state0,
                                              float* __restrict__ ws,
                                              float* __restrict__ out) {
  const int tid = threadIdx.x;
  const int b = tid >> 4, n = tid & 15;
  float Tcol[16];
#pragma unroll
  for (int m = 0; m < 16; ++m) Tcol[m] = ws[WS_P + LCH * 256 + m * 16 + n];
  float carry = state0[b * 16 + n];
  for (int c = 0; c < NCHB; ++c) {
    ws[WS_CARRY + (b * NCHB + c) * SDIM + n] = carry;
    float nc = ws[WS_SL + (b * SEQ + c * LCH + (LCH - 1)) * SDIM + n];
#pragma unroll
    for (int m = 0; m < 16; ++m) nc += __shfl(carry, m, 16) * Tcol[m];
    carry = nc;
  }
  out[ROWS * HID + b * 16 + n] = carry;  // final_state
}

// ===========================================================================
// Kernel 5: s_t = Sl_t + carry_c * P_{j+1}  (parallel carry correction)
// ===========================================================================
__global__ __launch_bounds__(256) void k_correct(float* __restrict__ ws) {
  const int g = blockIdx.x * 256 + threadIdx.x;
  const int row = g >> 4, n = g & 15;
  const int j2 = row & (LCH - 1), c = row >> 5;  // row%32 == t%32 (4096%32==0)
  float acc = ws[WS_SL + g];
  const float* __restrict__ P  = ws + WS_P + (j2 + 1) * 256;
  const float* __restrict__ cr = ws + WS_CARRY + c * SDIM;
#pragma unroll
  for (int m = 0; m < 16; ++m) acc += cr[m] * P[m * 16 + n];
  ws[WS_S + g] = acc;
}

// ===========================================================================
// Kernel 6: out = states @ C^T  (M=16384, K=16, N=1024) with f32 WMMA.
// Wave owns one 16-row tile and 8 h-tiles; A-fragments loaded once, 4 chained
// 16x16x4 WMMAs cover K=16 per tile.
// ===========================================================================
__global__ __launch_bounds__(256) void k_out_gemm(const float* __restrict__ Cmat,
                                                  const float* __restrict__ ws,
                                                  float* __restrict__ out) {
  const int tid = threadIdx.x;
  const int wglobal = blockIdx.x * 8 + (tid >> 5);
  const int rt = wglobal >> 3, hg = wglobal & 7;
  const int lane = tid & 31, M = lane & 15, half = lane >> 4;
  const float* __restrict__ S = ws + WS_S;

  v2f a[4];
#pragma unroll
  for (int k4 = 0; k4 < 4; ++k4)
    a[k4] = *(const v2f*)&S[(rt * 16 + M) * SDIM + k4 * 4 + 2 * half];

#pragma unroll
  for (int ht = 0; ht < 8; ++ht) {
    const int hbase = (hg * 8 + ht) * 16;
    v8f acc = {0.f, 0.f, 0.f, 0.f, 0.f, 0.f, 0.f, 0.f};
#pragma unroll
    for (int k4 = 0; k4 < 4; ++k4) {
      v2f b = *(const v2f*)&Cmat[(hbase + M) * SDIM + k4 * 4 + 2 * half];
      acc = wmma_f32_16x16x4(a[k4], b, acc);
    }
#pragma unroll
    for (int v = 0; v < 8; ++v)
      out[(rt * 16 + v + 8 * half) * HID + hbase + M] = acc[v];
  }
}

// ===========================================================================
extern "C" void kernel_launch(void* const* d_in, const int* in_sizes, int n_in,
                              void* d_out, int out_size, void* d_ws,
                              size_t ws_size, hipStream_t stream) {
  const float* x   = (const float*)d_in[0];  // [4,4096,1024]
  const float* st0 = (const float*)d_in[1];  // [4,16]
  const float* A   = (const float*)d_in[2];  // [16,16]
  const float* B   = (const float*)d_in[3];  // [16,1024]
  const float* C   = (const float*)d_in[4];  // [1024,16]
  const float* ld  = (const float*)d_in[5];  // [1024]
  float* out = (float*)d_out;                // [4,4096,1024] ++ [4,16]
  float* ws  = (float*)d_ws;                 // needs ~3.3 MB f32 scratch

  k_setup<<<1, 256, 0, stream>>>(A, B, ld, ws);
  k_u_gemm<<<ROWS / 128, 256, 0, stream>>>(x, ws);
  k_scan_local<<<NCH / 16, 256, 0, stream>>>(ws);
  k_carry<<<1, 64, 0, stream>>>(st0, ws, out);
  k_correct<<<(ROWS * SDIM) / 256, 256, 0, stream>>>(ws);
  k_out_gemm<<<(ROWS / 16), 256, 0, stream>>>(C, ws, out);
}